// Conv2d_91311004713559
// MI455X (gfx1250) — compile-verified
//
#include <hip/hip_runtime.h>
#include <hip/hip_bf16.h>

// Problem constants (match reference)
#define IN_CH   16
#define OUT_CH  16
#define HH      32
#define WW      32
#define H_OUT   30
#define W_OUT   30
#define POS     (H_OUT * W_OUT)          // 900
#define TPP     (IN_CH * 3 * 3)          // 144 tables per output pixel
#define NPIX    (OUT_CH * POS)           // 14400 output "pixels" (oc,pos)
#define BATCH   16
#define XELEMS  (IN_CH * HH * WW)        // 16384 elements per batch image

typedef __attribute__((ext_vector_type(2))) float v2f;
typedef __attribute__((ext_vector_type(4))) float v4f;
typedef __attribute__((ext_vector_type(8))) float v8f;
typedef __attribute__((ext_vector_type(4))) int   v4i;

// ---------------------------------------------------------------------------
// Prologue: transpose x [B=16, 16384] -> xT [16384, 16] so that the random
// gathers (lanes = batches) hit one contiguous 64B line per half-wave.
// ---------------------------------------------------------------------------
__global__ __launch_bounds__(256) void transpose_x_kernel(
    const float* __restrict__ x, float* __restrict__ xT) {
  int j = blockIdx.x * 256 + threadIdx.x;      // 0 .. 262143
  int b   = j & 15;
  int lin = j >> 4;
  xT[j] = x[b * XELEMS + lin];                 // coalesced store, gathered read (1MB total)
}

// ---------------------------------------------------------------------------
// Main kernel: 1 wave per output pixel, lane = (kpart<<4)|batch.
// Per chunk of 4 tables: half-wave kpart evaluates tables {2k, 2k+1} of the
// chunk for its batch; results form the A matrix of V_WMMA_F32_16X16X4_F32
// (M = batch, K = table-in-chunk). B = all-ones reduces across K; C=D
// accumulates across the 36 chunks -> D[m, n] = sum over 144 tables.
// ---------------------------------------------------------------------------
__global__ __launch_bounds__(256) void lutconv_wmma_kernel(
    const float* __restrict__ xT,       // [16384, 16]
    const int*   __restrict__ mask,     // [TABLES*2, 3] int32
    const float* __restrict__ weight,   // [TABLES, 4]
    float* __restrict__ out)            // [16, 14400]
{
  const int wave  = threadIdx.x >> 5;
  const int lane  = threadIdx.x & 31;
  const int p     = blockIdx.x * 8 + wave;     // pixel (oc*POS + pos), 0..14399
  const int b     = lane & 15;                 // batch
  const int kpart = lane >> 4;                 // half-wave: tables {0,1} or {2,3} of chunk

  const long tblBase = (long)p * TPP;                  // first table of this pixel
  const int* __restrict__ mbase = mask   + tblBase * 6; // 6 ints per table (2 rows x 3)
  const float* __restrict__ wbase = weight + tblBase * 4;

  const v2f ones = {1.0f, 1.0f};
  v8f acc = {};                                 // C/D accumulator (16x16 f32)

  for (int c = 0; c < TPP / 4; ++c) {           // 36 chunks of 4 tables
    const int t0 = 4 * c + 2 * kpart;           // first of this half-wave's 2 tables
    // mask rows for tables t0, t0+1: 12 ints = 48B, 16B-aligned (t0 even)
    const v4i* mp = (const v4i*)(mbase + 6 * t0);
    const v4i m0 = mp[0];
    const v4i m1 = mp[1];
    const v4i m2 = mp[2];
    const int lin00 = m0.x * 1024 + m0.y * 32 + m0.z;   // table t0,   k=0
    const int lin01 = m0.w * 1024 + m1.x * 32 + m1.y;   // table t0,   k=1
    const int lin10 = m1.z * 1024 + m1.w * 32 + m2.x;   // table t0+1, k=0
    const int lin11 = m2.y * 1024 + m2.z * 32 + m2.w;   // table t0+1, k=1

    // gathers: half-wave-uniform lin, lane-consecutive batch -> 64B line each
    const float x00 = xT[lin00 * 16 + b];
    const float x01 = xT[lin01 * 16 + b];
    const float x10 = xT[lin10 * 16 + b];
    const float x11 = xT[lin11 * 16 + b];

    const v4f* wp = (const v4f*)(wbase + 4 * t0);
    const v4f w0 = wp[0];
    const v4f w1 = wp[1];

    // table t0: P = ((am*w0+ap*w1)*bm + (am*w2+ap*w3)*bp), a*=f(x0), b*=f(x1)
    float am = fmaf(x00, -0.5f, 0.5f);
    float ap = fmaf(x00,  0.5f, 0.5f);
    float bm = fmaf(x01, -0.5f, 0.5f);
    float bp = fmaf(x01,  0.5f, 0.5f);
    float t1 = fmaf(ap, w0.y, am * w0.x);
    float t2 = fmaf(ap, w0.w, am * w0.z);
    const float P0 = fmaf(t2, bp, t1 * bm);

    // table t0+1
    am = fmaf(x10, -0.5f, 0.5f);
    ap = fmaf(x10,  0.5f, 0.5f);
    bm = fmaf(x11, -0.5f, 0.5f);
    bp = fmaf(x11,  0.5f, 0.5f);
    t1 = fmaf(ap, w1.y, am * w1.x);
    t2 = fmaf(ap, w1.w, am * w1.z);
    const float P1 = fmaf(t2, bp, t1 * bm);

    // A-matrix 16x4 f32 layout: VGPR0 = K=0 (lanes 0-15) / K=2 (lanes 16-31),
    // VGPR1 = K=1 / K=3.  M = batch = lane (mod 16).
    v2f a;
    a.x = P0;
    a.y = P1;
    // D = A x Ones + C : accumulates per-batch table sums into every column.
    acc = __builtin_amdgcn_wmma_f32_16x16x4_f32(
        /*neg_a=*/false, a, /*neg_b=*/false, ones,
        /*c_mod=*/(short)0, acc, /*reuse_a=*/false, /*reuse_b=*/false);
  }

  // Extract column N=0: lane 0 holds M=0..7 in acc[0..7], lane 16 holds M=8..15.
  // out flat layout: [B, OUT_CH, H_OUT, W_OUT] -> index b*NPIX + p.
  if (lane == 0) {
#pragma unroll
    for (int r = 0; r < 8; ++r) out[r * NPIX + p] = acc[r];
  } else if (lane == 16) {
#pragma unroll
    for (int r = 0; r < 8; ++r) out[(r + 8) * NPIX + p] = acc[r];
  }
}

// ---------------------------------------------------------------------------
extern "C" void kernel_launch(void* const* d_in, const int* in_sizes, int n_in,
                              void* d_out, int out_size, void* d_ws, size_t ws_size,
                              hipStream_t stream) {
  const float* x      = (const float*)d_in[0];   // [16,16,32,32] f32
  const int*   maskp  = (const int*)d_in[1];     // [TABLES*2, 3] i32
  const float* weight = (const float*)d_in[2];   // [TABLES, 4] f32
  float* out = (float*)d_out;                    // [16,16,30,30] f32
  float* xT  = (float*)d_ws;                     // 1 MB scratch: [16384,16] f32

  transpose_x_kernel<<<(BATCH * XELEMS) / 256, 256, 0, stream>>>(x, xT);
  lutconv_wmma_kernel<<<NPIX / 8, 256, 0, stream>>>(xT, maskp, weight, out);
}